// RotMat_6863357739349
// MI455X (gfx1250) — compile-verified
//
#include <hip/hip_runtime.h>

// ---------------------------------------------------------------------------
// Problem: X_out = (G_254 ... G_0) X, N=256 rows, B=8192 cols, fp32.
// Strategy:
//   Kernel 1: build R = G_254...G_0 (256x256) from the closed-form
//             round-robin schedule; 8 blocks each own a 32-column slice of R
//             in LDS (columns evolve independently under row rotations).
//   Kernel 2: Out = R * X as a WMMA fp32 GEMM (v_wmma_f32_16x16x4_f32),
//             HBM-bound: ~16MB traffic, R stays hot in 192MB L2.
//             X block staged TRANSPOSED in LDS so each B fragment is one
//             aligned ds_load_b64 (no VALU shuffles, conflict-free banks).
// ---------------------------------------------------------------------------

typedef __attribute__((ext_vector_type(2))) float v2f;
typedef __attribute__((ext_vector_type(8))) float v8f;

#define NROT 256
#define NCOL 8192
#define MOD  255       // NROT - 1
#define TSTR 260       // transposed-LDS row stride (dwords); 260%64=4

// ---------------------------------------------------------------------------
// Kernel 1: build R. grid = 8 blocks x 256 threads, dyn LDS = 256*32*4 = 32KB.
// Block b owns columns [32b, 32b+32) of R. 2 threads per rotation pair,
// each thread handles 16 consecutive floats (4 x float4) of the slice row.
// Schedule (circle method) closed form, round r:
//   pair 0 : (0, ((N-2-r) mod 255)+1)
//   pair k : i = ((k-1-r) mod 255)+1 , j = ((N-2-k-r) mod 255)+1
// theta index t = i*N - i(i+1)/2 + (j-i-1)  (after ordering i<j).
// ---------------------------------------------------------------------------
__global__ __launch_bounds__(256) void RotMat_build_R(
    const float* __restrict__ thetas, float* __restrict__ Rout) {
  extern __shared__ float Rs[];                 // [256][32] slice, stride 32
  const int tid = threadIdx.x;                  // 0..255
  const int c0  = blockIdx.x * 32;              // column slice base

  // identity slice
  for (int idx = tid; idx < NROT * 32; idx += 256) {
    int row = idx >> 5, col = idx & 31;
    Rs[idx] = (row == (c0 + col)) ? 1.0f : 0.0f;
  }
  __syncthreads();

  const int pair = tid >> 1;                    // 0..127
  const int sub  = (tid & 1) * 16;              // 0 or 16 (float offset)

  for (int r = 0; r < NROT - 1; ++r) {
    int i, j;
    if (pair == 0) {
      i = 0;
      j = ((NROT - 2 - r) % MOD) + 1;           // arg >= 0 here
    } else {
      int ai = (pair - 1 - r) % MOD;            // may be negative in C
      int aj = (NROT - 2 - pair - r) % MOD;
      i = ((ai + MOD) % MOD) + 1;
      j = ((aj + MOD) % MOD) + 1;
    }
    if (i > j) { int t = i; i = j; j = t; }
    const int tix = i * NROT - (i * (i + 1)) / 2 + (j - i - 1);
    const float th = thetas[tix];
    float sn, cs;
    sincosf(th, &sn, &cs);

    float4* ri = (float4*)&Rs[i * 32 + sub];
    float4* rj = (float4*)&Rs[j * 32 + sub];
#pragma unroll
    for (int q = 0; q < 4; ++q) {
      float4 a = ri[q], b = rj[q];
      float4 na, nb;
      na.x = cs * a.x - sn * b.x;  nb.x = sn * a.x + cs * b.x;
      na.y = cs * a.y - sn * b.y;  nb.y = sn * a.y + cs * b.y;
      na.z = cs * a.z - sn * b.z;  nb.z = sn * a.z + cs * b.z;
      na.w = cs * a.w - sn * b.w;  nb.w = sn * a.w + cs * b.w;
      ri[q] = na;  rj[q] = nb;
    }
    __syncthreads();
  }

  // write slice to global R (row-major 256x256)
  for (int idx = tid; idx < NROT * 32; idx += 256) {
    int row = idx >> 5, col = idx & 31;
    Rout[row * NROT + c0 + col] = Rs[idx];
  }
}

// ---------------------------------------------------------------------------
// Kernel 2: Out = R * X via V_WMMA_F32_16X16X4_F32.
// grid = 256 blocks (32-col blocks of X), 512 threads = 16 waves.
// Wave w computes rows [16w,16w+16) for the whole 32-col block (2 N-tiles).
// X block staged transposed in LDS: Xs[col*260 + row]; a B fragment for
// (k, col) is the contiguous pair (Xs[col*260+k+2h], Xs[col*260+k+2h+1])
// -> single ds_load_b64, banks (4*col + k + 2h) mod 64: all 64 banks hit
// exactly once per wave load (conflict-free).
// Fragment layouts (ISA 7.12.2, wave32):
//   A 16x4 : lane L, vgpr v -> M = L%16,        K = 2*(L/16)+v
//   B 4x16 : lane L, vgpr v -> K = 2*(L/16)+v,  N = L%16
//   C/D    : lane L, vgpr v -> M = v+8*(L/16),  N = L%16
// ---------------------------------------------------------------------------
__global__ __launch_bounds__(512) void RotMat_gemm(
    const float* __restrict__ R, const float* __restrict__ X,
    float* __restrict__ Out) {
  extern __shared__ float Xs[];                 // [32 cols][260] transposed
  const int n0  = blockIdx.x * 32;
  const int tid = threadIdx.x;

  // stage 256x32 X block into LDS, transposed (coalesced b128 global reads)
  for (int idx = tid; idx < NROT * 8; idx += 512) {
    int row = idx >> 3, c4 = (idx & 7) * 4;
    float4 val = *(const float4*)(X + row * NCOL + n0 + c4);
    Xs[(c4 + 0) * TSTR + row] = val.x;
    Xs[(c4 + 1) * TSTR + row] = val.y;
    Xs[(c4 + 2) * TSTR + row] = val.z;
    Xs[(c4 + 3) * TSTR + row] = val.w;
  }
  __syncthreads();

  const int wave = tid >> 5;
  const int lane = tid & 31;
  const int hl   = lane >> 4;                   // half-wave 0/1
  const int lq   = lane & 15;
  const int m0   = wave * 16;

  v8f acc0 = {};
  v8f acc1 = {};
  const float* arow  = R + (m0 + lq) * NROT + 2 * hl;   // A: per-lane row of R
  const float* brow0 = Xs + lq * TSTR + 2 * hl;         // B tile 0: col = lq
  const float* brow1 = Xs + (16 + lq) * TSTR + 2 * hl;  // B tile 1: col = 16+lq

#pragma unroll 4
  for (int k = 0; k < NROT; k += 4) {
    v2f a  = *(const v2f*)(arow  + k);          // global_load_b64 (L2-hot R)
    v2f b0 = *(const v2f*)(brow0 + k);          // ds_load_b64, K=(2h, 2h+1)
    v2f b1 = *(const v2f*)(brow1 + k);          // ds_load_b64
    acc0 = __builtin_amdgcn_wmma_f32_16x16x4_f32(
        false, a, false, b0, (short)0, acc0, false, false);
    acc1 = __builtin_amdgcn_wmma_f32_16x16x4_f32(
        false, a, false, b1, (short)0, acc1, false, false);
  }

  // store D tiles (coalesced 64B per half-wave)
#pragma unroll
  for (int v = 0; v < 8; ++v) {
    const int row = m0 + v + 8 * hl;
    Out[row * NCOL + n0 + lq]      = acc0[v];
    Out[row * NCOL + n0 + 16 + lq] = acc1[v];
  }
}

// ---------------------------------------------------------------------------
extern "C" void kernel_launch(void* const* d_in, const int* in_sizes, int n_in,
                              void* d_out, int out_size, void* d_ws, size_t ws_size,
                              hipStream_t stream) {
  const float* X      = (const float*)d_in[0];   // [256, 8192] fp32
  const float* thetas = (const float*)d_in[1];   // [32640] fp32
  float*       Out    = (float*)d_out;           // [256, 8192] fp32
  float*       Rws    = (float*)d_ws;            // 256*256 fp32 = 256KB scratch

  RotMat_build_R<<<8, 256, NROT * 32 * sizeof(float), stream>>>(thetas, Rws);
  RotMat_gemm<<<NCOL / 32, 512, 32 * TSTR * sizeof(float), stream>>>(Rws, X, Out);
}